// GCN_42021960024156
// MI455X (gfx1250) — compile-verified
//
#include <hip/hip_runtime.h>
#include <hip/hip_bf16.h>

#define N_NODES 50000
#define N_EDGES 800000
#define F1 128
#define F2 47
#define F2P 48

typedef __attribute__((ext_vector_type(2))) float v2f;
typedef __attribute__((ext_vector_type(8))) float v8f;

// Native fp32 atomic add (GLOBAL_ATOMIC_ADD_F32, no-return) instead of a
// possible CAS loop from generic atomicAdd under safe-fp-atomics.
__device__ __forceinline__ void atom_add_f32(float* p, float v) {
    unsafeAtomicAdd(p, v);
}

// ---------------------------------------------------------------- utilities

__global__ __launch_bounds__(256) void gcn_zero(float* __restrict__ p, unsigned n) {
    unsigned i = blockIdx.x * 256u + threadIdx.x;
    if (i < n) p[i] = 0.0f;
}

__global__ __launch_bounds__(256) void gcn_degree(const int* __restrict__ src,
                                                  const int* __restrict__ dst,
                                                  float* __restrict__ deg_s,
                                                  float* __restrict__ deg_d) {
    unsigned e = blockIdx.x * 256u + threadIdx.x;
    if (e >= N_EDGES) return;
    atom_add_f32(&deg_s[src[e]], 1.0f);
    atom_add_f32(&deg_d[dst[e]], 1.0f);
}

__global__ __launch_bounds__(256) void gcn_norm(float* __restrict__ d, unsigned n) {
    unsigned i = blockIdx.x * 256u + threadIdx.x;
    if (i >= n) return;
    d[i] = rsqrtf(fmaxf(d[i], 1.0f));
}

// ------------------------------------------------- GEMM1: H = (X*ns) @ W1
// M=50000 (3125 tiles of 16, exact), N=128 (8 waves x 16), K=128 (32 x K4 WMMA)

__global__ __launch_bounds__(256) void gcn_gemm1(const float* __restrict__ X,
                                                 const float* __restrict__ W,
                                                 const float* __restrict__ ns,
                                                 float* __restrict__ H) {
    __shared__ float As[16][132];             // pad 132: banks spread (4m+k)
    const int m0 = blockIdx.x * 16;
    // stage A strip, fuse norm_src scaling (2048 elements / 256 threads)
    for (int i = threadIdx.x; i < 16 * F1; i += 256) {
        int r = i >> 7, c = i & 127;
        int m = m0 + r;
        As[r][c] = X[(size_t)m * F1 + c] * ns[m];
    }
    __syncthreads();

    const int lane  = threadIdx.x & 31;
    const int wave  = threadIdx.x >> 5;       // N tile 0..7
    const int mrow  = lane & 15;
    const int khalf = lane >> 4;              // 0: K pair {0,1}, 1: {2,3}
    const int n     = (wave << 4) + (lane & 15);

    v8f acc = {};
    for (int k0 = 0; k0 < F1; k0 += 4) {
        const int ka = k0 + 2 * khalf;
        v2f a; a.x = As[mrow][ka];            a.y = As[mrow][ka + 1];
        v2f b; b.x = W[(size_t)ka * F1 + n];  b.y = W[(size_t)(ka + 1) * F1 + n];
        acc = __builtin_amdgcn_wmma_f32_16x16x4_f32(
            false, a, false, b, (short)0, acc, false, false);
    }
    #pragma unroll
    for (int r = 0; r < 8; ++r) {
        int m = m0 + 8 * khalf + r;
        H[(size_t)m * F1 + n] = acc[r];
    }
}

// ------------------------------------------------- GEMM2: H2 = X2 @ W2 (N pad 47->48)

__global__ __launch_bounds__(96) void gcn_gemm2(const float* __restrict__ X2,
                                                const float* __restrict__ W2,
                                                float* __restrict__ H2) {
    __shared__ float As[16][132];
    const int m0 = blockIdx.x * 16;
    for (int i = threadIdx.x; i < 16 * F1; i += 96) {
        int r = i >> 7, c = i & 127;
        As[r][c] = X2[(size_t)(m0 + r) * F1 + c];
    }
    __syncthreads();

    const int lane  = threadIdx.x & 31;
    const int wave  = threadIdx.x >> 5;       // N tile 0..2
    const int mrow  = lane & 15;
    const int khalf = lane >> 4;
    const int n     = (wave << 4) + (lane & 15);
    const bool valid = (n < F2);

    v8f acc = {};
    for (int k0 = 0; k0 < F1; k0 += 4) {
        const int ka = k0 + 2 * khalf;
        v2f a; a.x = As[mrow][ka]; a.y = As[mrow][ka + 1];
        v2f b;
        b.x = valid ? W2[(size_t)ka * F2 + n] : 0.0f;
        b.y = valid ? W2[(size_t)(ka + 1) * F2 + n] : 0.0f;
        acc = __builtin_amdgcn_wmma_f32_16x16x4_f32(
            false, a, false, b, (short)0, acc, false, false);
    }
    #pragma unroll
    for (int r = 0; r < 8; ++r) {
        int m = m0 + 8 * khalf + r;
        H2[(size_t)m * F2P + n] = acc[r];     // col 47 holds zeros (B col zero)
    }
}

// ------------------------------------------------- scatter-add over edges

template <int F, int LPE>
__global__ __launch_bounds__(256) void gcn_scatter(const float* __restrict__ H,
                                                   const int* __restrict__ src,
                                                   const int* __restrict__ dst,
                                                   float* __restrict__ AGG) {
    const unsigned total = (unsigned)N_EDGES * (unsigned)LPE;
    unsigned t = blockIdx.x * 256u + threadIdx.x;
    if (t >= total) return;
    unsigned e = t / (unsigned)LPE;           // LPE is compile-time constant
    unsigned f = (t - e * (unsigned)LPE) * 4u;
    int s = src[e], d = dst[e];
    const float4 v = *reinterpret_cast<const float4*>(H + (size_t)s * F + f);
    float* o = AGG + (size_t)d * F + f;
    atom_add_f32(o + 0, v.x);
    atom_add_f32(o + 1, v.y);
    atom_add_f32(o + 2, v.z);
    atom_add_f32(o + 3, v.w);
}

// ------------------------------------------------- epilogues

// xs2 = relu(agg1 * nd + b1) * ns   (in place; folds layer-2 source scaling)
__global__ __launch_bounds__(256) void gcn_relu_scale(float* __restrict__ buf,
                                                      const float* __restrict__ nd,
                                                      const float* __restrict__ ns,
                                                      const float* __restrict__ b1) {
    unsigned i = blockIdx.x * 256u + threadIdx.x;
    if (i >= (unsigned)N_NODES * F1) return;
    unsigned n = i >> 7, f = i & 127;
    float v = fmaf(buf[i], nd[n], b1[f]);
    buf[i] = fmaxf(v, 0.0f) * ns[n];
}

// out = agg2 * nd + b2  (drop pad column)
__global__ __launch_bounds__(256) void gcn_final(const float* __restrict__ agg2,
                                                 const float* __restrict__ nd,
                                                 const float* __restrict__ b2,
                                                 float* __restrict__ out) {
    unsigned i = blockIdx.x * 256u + threadIdx.x;
    if (i >= (unsigned)N_NODES * F2) return;
    unsigned n = i / F2;
    unsigned c = i - n * F2;
    out[i] = fmaf(agg2[(size_t)n * F2P + c], nd[n], b2[c]);
}

// ---------------------------------------------------------------- launch

extern "C" void kernel_launch(void* const* d_in, const int* in_sizes, int n_in,
                              void* d_out, int out_size, void* d_ws, size_t ws_size,
                              hipStream_t stream) {
    const float* feat = (const float*)d_in[0];
    const int*   eidx = (const int*)d_in[1];
    const float* W1   = (const float*)d_in[2];
    const float* b1   = (const float*)d_in[3];
    const float* W2   = (const float*)d_in[4];
    const float* b2   = (const float*)d_in[5];
    const int* src = eidx;
    const int* dst = eidx + N_EDGES;
    float* out = (float*)d_out;
    float* ws  = (float*)d_ws;

    const size_t NS = 50048;                       // aligned node-vector slot
    float* norm_src = ws;                          // NS floats
    float* norm_dst = ws + NS;                     // NS floats
    float* bufA = ws + 2 * NS;                     // 6.4M floats: h, then h2
    float* bufB = bufA + (size_t)N_NODES * F1;     // 6.4M floats: agg1/xs2, then agg2

    const unsigned NB = (unsigned)N_NODES * F1;    // 6,400,000
    const unsigned NB2 = (unsigned)N_NODES * F2P;  // 2,400,000

    // degrees -> norms
    gcn_zero<<<(unsigned)(2 * NS + 255) / 256, 256, 0, stream>>>(ws, (unsigned)(2 * NS));
    gcn_degree<<<(N_EDGES + 255) / 256, 256, 0, stream>>>(src, dst, norm_src, norm_dst);
    gcn_norm<<<(unsigned)(2 * NS + 255) / 256, 256, 0, stream>>>(ws, (unsigned)(2 * NS));

    // layer 1
    gcn_gemm1<<<N_NODES / 16, 256, 0, stream>>>(feat, W1, norm_src, bufA);
    gcn_zero<<<(NB + 255) / 256, 256, 0, stream>>>(bufB, NB);
    gcn_scatter<F1, 32><<<(N_EDGES * 32u + 255) / 256, 256, 0, stream>>>(bufA, src, dst, bufB);
    gcn_relu_scale<<<(NB + 255) / 256, 256, 0, stream>>>(bufB, norm_dst, norm_src, b1);

    // layer 2
    gcn_gemm2<<<N_NODES / 16, 96, 0, stream>>>(bufB, W2, bufA);
    gcn_zero<<<(NB2 + 255) / 256, 256, 0, stream>>>(bufB, NB2);
    gcn_scatter<F2P, 12><<<(N_EDGES * 12u + 255) / 256, 256, 0, stream>>>(bufA, src, dst, bufB);
    gcn_final<<<((unsigned)N_NODES * F2 + 255) / 256, 256, 0, stream>>>(bufB, norm_dst, b2, out);
}